// GraphConvolutionalNetwork_2396591751687
// MI455X (gfx1250) — compile-verified
//
#include <hip/hip_runtime.h>
#include <hip/hip_bf16.h>
#include <math.h>

// ---------------------------------------------------------------------------
// GCN forward: o = sigmoid( L @ (relu( L @ (x@W1) + b1 ) @ W2) + b2 )
//   N = 100000, F = 512, H = 16, O = 64, NNZ = 3.2M (COO, unsorted, dup rows)
// MI455X strategy (wave32, fp32 WMMA, async LDS DMA):
//   gemm1 : x@W1 with V_WMMA_F32_16X16X4_F32.
//           - W1^T staged once per block in LDS (conflict-free b64 frags)
//           - x tiles double-buffered via GLOBAL_LOAD_ASYNC_TO_LDS_B128
//             (ASYNCcnt pipelined against the WMMA stream)
//   spmm  : per-edge float4 gather + hardware fp32 L2 atomics scatter
//   gemm2 : h@W2 with WMMA, relu fused into A-fragment load
// ---------------------------------------------------------------------------

typedef __attribute__((ext_vector_type(2))) float v2f;
typedef __attribute__((ext_vector_type(8))) float v8f;

#define N_NODES 100000
#define N_FEAT  512
#define HIDDEN  16
#define OUT_DIM 64
#define TILES   (N_NODES / 16)   // 6250, exact
#define KCHUNK  128
#define NCHUNKS (N_FEAT / KCHUNK)
#define LDS_STRIDE 132           // 128 + 4 pad -> conflict-free frag reads
#define W1T_STRIDE 516           // 512 + 4 pad

// ------------------------- GEMM1: xw1 = x @ W1 ------------------------------
__global__ __launch_bounds__(256)
void gemm1_wmma(const float* __restrict__ x, const float* __restrict__ W1,
                float* __restrict__ xw1) {
    __shared__ float ldsX[8][2][16 * LDS_STRIDE];   // per-wave double buffers
    __shared__ float ldsW1T[16 * W1T_STRIDE];       // W1 transposed, block-shared

    const int wid  = threadIdx.x >> 5;
    const int lane = threadIdx.x & 31;
    const int tile = blockIdx.x * 8 + wid;
    const bool active = (tile < TILES);

    // Stage W1^T into LDS once (coalesced global read, conflict-free scatter)
    for (int idx = threadIdx.x; idx < N_FEAT * HIDDEN; idx += 256) {
        const int k = idx >> 4;
        const int n = idx & 15;
        ldsW1T[n * W1T_STRIDE + k] = W1[idx];
    }
    __syncthreads();
    if (!active) return;

    const int row0 = tile * 16;
    const int m    = lane & 15;          // row within tile (A) / col (B,D)
    const int half = lane >> 4;          // 0: K,K+1   1: K+2,K+3

    // Low 32 bits of a flat shared-memory address are the LDS byte offset.
    const unsigned ldsB0 = (unsigned)(uintptr_t)&ldsX[wid][0][0];
    const unsigned ldsB1 = (unsigned)(uintptr_t)&ldsX[wid][1][0];

    // Async DMA one 16x128 chunk of x into an LDS buffer (16 x B128 per lane
    // column-slot; each instruction moves a contiguous 512B row segment).
    auto stage = [&](int kc, unsigned ldsbase) {
        // WAR guard: prior ds_loads from this buffer must have drained.
        asm volatile("s_wait_dscnt 0x0" ::: "memory");
        const unsigned lcol = (unsigned)(lane << 4);         // lane*16 bytes
        for (int r = 0; r < 16; ++r) {
            const unsigned loff = ldsbase + (unsigned)(r * LDS_STRIDE * 4) + lcol;
            const unsigned goff = (unsigned)(((row0 + r) * N_FEAT + kc) * 4) + lcol;
            asm volatile("global_load_async_to_lds_b128 %0, %1, %2"
                         :: "v"(loff), "v"(goff), "s"(x) : "memory");
        }
    };

    v8f acc = {};
    stage(0, ldsB0);
    for (int c = 0; c < NCHUNKS; ++c) {
        if (c + 1 < NCHUNKS) {
            stage((c + 1) * KCHUNK, ((c + 1) & 1) ? ldsB1 : ldsB0);
            // oldest 16 async ops (current chunk) complete, 16 still in flight
            asm volatile("s_wait_asynccnt 0x10" ::: "memory");
        } else {
            asm volatile("s_wait_asynccnt 0x0" ::: "memory");
        }
        const float* xb = &ldsX[wid][c & 1][0];
        const int kc = c * KCHUNK;
        for (int k = 0; k < KCHUNK; k += 4) {
            const int kb = k + (half << 1);
            v2f a = *(const v2f*)&xb[m * LDS_STRIDE + kb];
            v2f b = *(const v2f*)&ldsW1T[m * W1T_STRIDE + kc + kb];
            acc = __builtin_amdgcn_wmma_f32_16x16x4_f32(
                false, a, false, b, (short)0, acc, false, false);
        }
    }
    // D layout: vgpr r holds M = r + half*8, N = lane&15
    for (int r = 0; r < 8; ++r) {
        const int mm = r + (half << 3);
        xw1[(size_t)(row0 + mm) * HIDDEN + m] = acc[r];
    }
}

// ------------------------- init buffers with bias ---------------------------
__global__ void init_bias(float* __restrict__ dst, const float* __restrict__ bias,
                          int n_rows, int dim) {
    const long long t = (long long)blockIdx.x * blockDim.x + threadIdx.x;
    const long long total = (long long)n_rows * dim;
    if (t >= total) return;
    dst[t] = bias[t % dim];
}

// --------------- SpMM scatter: dst[row] += val * src[col] -------------------
__global__ void spmm_scatter16(const int* __restrict__ rows, const int* __restrict__ cols,
                               const float* __restrict__ vals,
                               const float* __restrict__ src, float* __restrict__ dst,
                               long long nnz) {
    const long long t = (long long)blockIdx.x * blockDim.x + threadIdx.x;
    if (t >= nnz * 4) return;
    const long long e = t >> 2;
    const int f4 = (int)(t & 3) << 2;
    const int r = rows[e], c = cols[e];
    const float v = vals[e];
    const float4 g = *(const float4*)(src + (size_t)c * 16 + f4);
    float* d = dst + (size_t)r * 16 + f4;
    unsafeAtomicAdd(d + 0, v * g.x);
    unsafeAtomicAdd(d + 1, v * g.y);
    unsafeAtomicAdd(d + 2, v * g.z);
    unsafeAtomicAdd(d + 3, v * g.w);
}

__global__ void spmm_scatter64(const int* __restrict__ rows, const int* __restrict__ cols,
                               const float* __restrict__ vals,
                               const float* __restrict__ src, float* __restrict__ dst,
                               long long nnz) {
    const long long t = (long long)blockIdx.x * blockDim.x + threadIdx.x;
    if (t >= nnz * 16) return;
    const long long e = t >> 4;
    const int f4 = (int)(t & 15) << 2;
    const int r = rows[e], c = cols[e];
    const float v = vals[e];
    const float4 g = *(const float4*)(src + (size_t)c * 64 + f4);
    float* d = dst + (size_t)r * 64 + f4;
    unsafeAtomicAdd(d + 0, v * g.x);
    unsafeAtomicAdd(d + 1, v * g.y);
    unsafeAtomicAdd(d + 2, v * g.z);
    unsafeAtomicAdd(d + 3, v * g.w);
}

// --------------- GEMM2: hw2 = relu(h) @ W2   (K=16, N=64) -------------------
__global__ __launch_bounds__(256)
void gemm2_wmma(const float* __restrict__ h, const float* __restrict__ W2,
                float* __restrict__ hw2) {
    const int wid  = threadIdx.x >> 5;
    const int lane = threadIdx.x & 31;
    const int tile = blockIdx.x * 8 + wid;
    if (tile >= TILES) return;

    const int row0 = tile * 16;
    const int m    = lane & 15;
    const int half = lane >> 4;

    v8f acc0 = {}, acc1 = {}, acc2 = {}, acc3 = {};
    for (int k = 0; k < HIDDEN; k += 4) {
        const int kb = k + (half << 1);
        // A fragment with fused relu (h is L2-resident: 6.4 MB)
        v2f a;
        a.x = fmaxf(h[(size_t)(row0 + m) * HIDDEN + kb    ], 0.0f);
        a.y = fmaxf(h[(size_t)(row0 + m) * HIDDEN + kb + 1], 0.0f);
        // B fragments for the four 16-col tiles of W2 [16 x 64]
        v2f b0, b1, b2, b3;
        b0.x = W2[(size_t)(kb    ) * OUT_DIM + m +  0];
        b0.y = W2[(size_t)(kb + 1) * OUT_DIM + m +  0];
        b1.x = W2[(size_t)(kb    ) * OUT_DIM + m + 16];
        b1.y = W2[(size_t)(kb + 1) * OUT_DIM + m + 16];
        b2.x = W2[(size_t)(kb    ) * OUT_DIM + m + 32];
        b2.y = W2[(size_t)(kb + 1) * OUT_DIM + m + 32];
        b3.x = W2[(size_t)(kb    ) * OUT_DIM + m + 48];
        b3.y = W2[(size_t)(kb + 1) * OUT_DIM + m + 48];
        acc0 = __builtin_amdgcn_wmma_f32_16x16x4_f32(false, a, false, b0, (short)0, acc0, false, false);
        acc1 = __builtin_amdgcn_wmma_f32_16x16x4_f32(false, a, false, b1, (short)0, acc1, false, false);
        acc2 = __builtin_amdgcn_wmma_f32_16x16x4_f32(false, a, false, b2, (short)0, acc2, false, false);
        acc3 = __builtin_amdgcn_wmma_f32_16x16x4_f32(false, a, false, b3, (short)0, acc3, false, false);
    }
    for (int r = 0; r < 8; ++r) {
        const int mm = r + (half << 3);
        float* o = hw2 + (size_t)(row0 + mm) * OUT_DIM + m;
        o[ 0] = acc0[r];
        o[16] = acc1[r];
        o[32] = acc2[r];
        o[48] = acc3[r];
    }
}

// ------------------------- sigmoid epilogue ---------------------------------
__global__ void sigmoid_inplace(float* __restrict__ o, long long n4) {
    const long long t = (long long)blockIdx.x * blockDim.x + threadIdx.x;
    if (t >= n4) return;
    float4 v = *(float4*)(o + t * 4);
    v.x = 1.0f / (1.0f + __expf(-v.x));
    v.y = 1.0f / (1.0f + __expf(-v.y));
    v.z = 1.0f / (1.0f + __expf(-v.z));
    v.w = 1.0f / (1.0f + __expf(-v.w));
    *(float4*)(o + t * 4) = v;
}

// ---------------------------------------------------------------------------
extern "C" void kernel_launch(void* const* d_in, const int* in_sizes, int n_in,
                              void* d_out, int out_size, void* d_ws, size_t ws_size,
                              hipStream_t stream) {
    const float* x        = (const float*)d_in[0];
    const int*   lap_rows = (const int*)  d_in[1];
    const int*   lap_cols = (const int*)  d_in[2];
    const float* lap_vals = (const float*)d_in[3];
    const float* W1       = (const float*)d_in[4];
    const float* b1       = (const float*)d_in[5];
    const float* W2       = (const float*)d_in[6];
    const float* b2       = (const float*)d_in[7];
    float*       out      = (float*)d_out;

    const long long nnz = in_sizes[1];

    // workspace layout (floats): xw1[N*16] | h[N*16] | hw2[N*64]  (38.4 MB)
    float* xw1 = (float*)d_ws;
    float* h   = xw1 + (size_t)N_NODES * HIDDEN;
    float* hw2 = h   + (size_t)N_NODES * HIDDEN;

    const int gemm_blocks = (TILES + 7) / 8;

    // layer 1
    gemm1_wmma<<<gemm_blocks, 256, 0, stream>>>(x, W1, xw1);
    {
        long long tot = (long long)N_NODES * HIDDEN;
        init_bias<<<(unsigned)((tot + 255) / 256), 256, 0, stream>>>(h, b1, N_NODES, HIDDEN);
    }
    {
        long long thr = nnz * 4;
        spmm_scatter16<<<(unsigned)((thr + 255) / 256), 256, 0, stream>>>(
            lap_rows, lap_cols, lap_vals, xw1, h, nnz);
    }

    // layer 2 (relu fused into gemm2 A-load)
    gemm2_wmma<<<gemm_blocks, 256, 0, stream>>>(h, W2, hw2);
    {
        long long tot = (long long)N_NODES * OUT_DIM;
        init_bias<<<(unsigned)((tot + 255) / 256), 256, 0, stream>>>(out, b2, N_NODES, OUT_DIM);
    }
    {
        long long thr = nnz * 16;
        spmm_scatter64<<<(unsigned)((thr + 255) / 256), 256, 0, stream>>>(
            lap_rows, lap_cols, lap_vals, hw2, out, nnz);
    }
    {
        long long n4 = (long long)N_NODES * OUT_DIM / 4;
        sigmoid_inplace<<<(unsigned)((n4 + 255) / 256), 256, 0, stream>>>(out, n4);
    }
}